// tableInterpolation_47227460387485
// MI455X (gfx1250) — compile-verified
//
#include <hip/hip_runtime.h>

// Bilinear table interpolation, memory-bound gather kernel for gfx1250.
// - NT hints on streaming input/output (protect L2-resident 4MB grid)
// - paired 8B gathers (tl,tr) and (bl,br), bottom pair at +W*4 bytes
// - 8 points/thread fully unrolled for memory-level parallelism
// - global_prefetch warm-up of the grid into L2

#define HGRID 1024
#define WGRID 1024

typedef float v4f __attribute__((ext_vector_type(4)));

// 8-byte pair with only 4-byte alignment guarantee (x0 is any column).
struct __attribute__((packed, aligned(4))) fpair { float lo, hi; };

__device__ __forceinline__ float bilerp1(float py, float px,
                                         float sy, float sx,
                                         float ylo, float xlo,
                                         const float* __restrict__ grid) {
    float qy = (py - ylo) * sy;
    float qx = (px - xlo) * sx;
    // floor clamped to [0, size-2]; alpha clamped to [0,1] (matches reference)
    float fy = fminf(fmaxf(floorf(qy), 0.0f), (float)(HGRID - 2));
    float fx = fminf(fmaxf(floorf(qx), 0.0f), (float)(WGRID - 2));
    float ay = fminf(fmaxf(qy - fy, 0.0f), 1.0f);
    float ax = fminf(fmaxf(qx - fx, 0.0f), 1.0f);
    int y0 = (int)fy;
    int x0 = (int)fx;
    const float* cell = grid + (y0 * WGRID + x0);
    fpair tp = *reinterpret_cast<const fpair*>(cell);          // tl, tr
    fpair bp = *reinterpret_cast<const fpair*>(cell + WGRID);  // bl, br (+4096B)
    float top = tp.lo + (tp.hi - tp.lo) * ax;
    float bot = bp.lo + (bp.hi - bp.lo) * ax;
    return top + (bot - top) * ay;
}

__global__ __launch_bounds__(256) void tableInterp_gfx1250_kernel(
    const float* __restrict__ in,      // [N,2] (x=row coord, v=col coord)
    const float* __restrict__ grid,    // [H,W]
    const float* __restrict__ bounds,  // [[ylo,yhi],[xlo,xhi]] flat
    float* __restrict__ out,           // [N]
    int npts) {
    const int t = blockIdx.x * blockDim.x + threadIdx.x;

    // Warm the 4MB grid into L2 (overlaps with first streaming loads).
    // 1024*1024*4B / 128B-lines = 32768 lines; one line per early thread.
    if (t < (HGRID * WGRID * 4) / 128) {
        __builtin_prefetch(grid + t * 32, 0, 1);  // -> global_prefetch_b8
    }

    const int base = t * 8;  // first point this thread handles
    if (base >= npts) return;

    // Uniform bounds (scalarized by the compiler).
    const float ylo = bounds[0], yhi = bounds[1];
    const float xlo = bounds[2], xhi = bounds[3];
    const float sy = (float)(HGRID - 1) / (yhi - ylo);
    const float sx = (float)(WGRID - 1) / (xhi - xlo);

    if (base + 8 <= npts) {
        // Fast path: 8 points = 16 input floats = 4x b128 NT loads.
        const v4f* in4 = reinterpret_cast<const v4f*>(in) + (size_t)t * 4;
        v4f p0 = __builtin_nontemporal_load(in4 + 0);
        v4f p1 = __builtin_nontemporal_load(in4 + 1);
        v4f p2 = __builtin_nontemporal_load(in4 + 2);
        v4f p3 = __builtin_nontemporal_load(in4 + 3);

        v4f r0, r1;
        r0[0] = bilerp1(p0[0], p0[1], sy, sx, ylo, xlo, grid);
        r0[1] = bilerp1(p0[2], p0[3], sy, sx, ylo, xlo, grid);
        r0[2] = bilerp1(p1[0], p1[1], sy, sx, ylo, xlo, grid);
        r0[3] = bilerp1(p1[2], p1[3], sy, sx, ylo, xlo, grid);
        r1[0] = bilerp1(p2[0], p2[1], sy, sx, ylo, xlo, grid);
        r1[1] = bilerp1(p2[2], p2[3], sy, sx, ylo, xlo, grid);
        r1[2] = bilerp1(p3[0], p3[1], sy, sx, ylo, xlo, grid);
        r1[3] = bilerp1(p3[2], p3[3], sy, sx, ylo, xlo, grid);

        v4f* out4 = reinterpret_cast<v4f*>(out) + (size_t)t * 2;
        __builtin_nontemporal_store(r0, out4 + 0);
        __builtin_nontemporal_store(r1, out4 + 1);
    } else {
        // Tail: scalar per-point loop.
        for (int i = base; i < npts; ++i) {
            float py = in[2 * i + 0];
            float px = in[2 * i + 1];
            out[i] = bilerp1(py, px, sy, sx, ylo, xlo, grid);
        }
    }
}

extern "C" void kernel_launch(void* const* d_in, const int* in_sizes, int n_in,
                              void* d_out, int out_size, void* d_ws, size_t ws_size,
                              hipStream_t stream) {
    const float* in     = (const float*)d_in[0];  // [N,2] float32
    const float* grid   = (const float*)d_in[1];  // [1024,1024] float32
    const float* bounds = (const float*)d_in[2];  // [2,2] float32
    float* out = (float*)d_out;                   // [N] float32 (flattened [1,N,1])

    const int npts = in_sizes[0] / 2;             // N
    const int threads_needed = (npts + 7) / 8;    // 8 points per thread
    const int block = 256;
    const int nblocks = (threads_needed + block - 1) / block;

    tableInterp_gfx1250_kernel<<<nblocks, block, 0, stream>>>(in, grid, bounds, out, npts);
}